// SIGN_24163486007758
// MI455X (gfx1250) — compile-verified
//
#include <hip/hip_runtime.h>
#include <math.h>

typedef __attribute__((ext_vector_type(2))) float v2f;
typedef __attribute__((ext_vector_type(8))) float v8f;

#define N_NODES 100000
#define N_EDGES 1600000
#define N_FEAT  256
#define N_HID   256
#define N_CLASS 40
#define LDSPITCH 260   // 256 + 4 pad: row*260 % 64 = 4*row -> 16 distinct banks

// ---------------------------------------------------------------- zero fill
__global__ void SIGN_zero_f32(float* __restrict__ p, long n) {
    long i = (long)blockIdx.x * blockDim.x + threadIdx.x;
    if (i < n) p[i] = 0.0f;
}

// ---------------------------------------------------------------- GEMM1: h0 = x @ W1
// grid.x = N_NODES/16 blocks, 256 threads = 8 waves.
// Block stages a 16x256 stripe of x in LDS; wave w computes cols [32w, 32w+32).
__global__ __launch_bounds__(256)
void SIGN_gemm1(const float* __restrict__ X, const float* __restrict__ W1,
                float* __restrict__ H0) {
    __shared__ float ldsA[16 * LDSPITCH];
    const int m0  = blockIdx.x << 4;
    const int tid = threadIdx.x;

    for (int i = tid; i < 16 * 256; i += 256) {
        int r = i >> 8, c = i & 255;
        ldsA[r * LDSPITCH + c] = X[(long)(m0 + r) * N_FEAT + c];
    }
    __syncthreads();

    const int lane = tid & 31;
    const int li = lane & 15;       // row (A) / col (B,C) within tile
    const int lh = lane >> 4;       // K-half select
    const int n0 = (tid >> 5) << 5; // wave * 32

    v8f c0 = {}; v8f c1 = {};
    for (int k = 0; k < N_FEAT; k += 4) {
        v2f a;
        a.x = ldsA[li * LDSPITCH + k + 2 * lh];
        a.y = ldsA[li * LDSPITCH + k + 2 * lh + 1];
        const float* Wp = W1 + (long)(k + 2 * lh) * N_HID;
        v2f b0, b1;
        b0.x = Wp[n0 + li];
        b0.y = Wp[N_HID + n0 + li];
        b1.x = Wp[n0 + 16 + li];
        b1.y = Wp[N_HID + n0 + 16 + li];
        c0 = __builtin_amdgcn_wmma_f32_16x16x4_f32(false, a, false, b0, (short)0, c0, false, false);
        c1 = __builtin_amdgcn_wmma_f32_16x16x4_f32(false, a, false, b1, (short)0, c1, false, false);
    }
#pragma unroll
    for (int r = 0; r < 8; ++r) {
        int row = m0 + r + 8 * lh;
        H0[(long)row * N_HID + n0 + li]      = c0[r];
        H0[(long)row * N_HID + n0 + 16 + li] = c1[r];
    }
}

// ---------------------------------------------------------------- SpMM (256 feats)
// one thread per (edge, float4 chunk): 64 chunks/edge
__global__ __launch_bounds__(256)
void SIGN_spmm256(const float* __restrict__ H, const int* __restrict__ src,
                  const int* __restrict__ dst, const float* __restrict__ w,
                  float* __restrict__ OUT) {
    long t = (long)blockIdx.x * blockDim.x + threadIdx.x;
    if (t >= (long)N_EDGES * 64) return;
    int e = (int)(t >> 6);
    int c = ((int)t & 63) << 2;
    int s = src[e], d = dst[e];
    float we = w[e];
    const float4 v = *(const float4*)(H + (long)s * N_HID + c);
    float* o = OUT + (long)d * N_HID + c;
    atomicAdd(o + 0, we * v.x);
    atomicAdd(o + 1, we * v.y);
    atomicAdd(o + 2, we * v.z);
    atomicAdd(o + 3, we * v.w);
}

// ---------------------------------------------------------------- GEMM2: l0 = relu(h1+b1) @ W2
// 96 threads = 3 waves; bias+ReLU fused into the LDS staging pass. N=40 padded to 48.
__global__ __launch_bounds__(96)
void SIGN_gemm2(const float* __restrict__ H1, const float* __restrict__ B1,
                const float* __restrict__ W2, float* __restrict__ L0) {
    __shared__ float ldsA[16 * LDSPITCH];
    const int m0  = blockIdx.x << 4;
    const int tid = threadIdx.x;

    for (int i = tid; i < 16 * 256; i += 96) {
        int r = i >> 8, c = i & 255;
        float v = H1[(long)(m0 + r) * N_HID + c] + B1[c];
        ldsA[r * LDSPITCH + c] = v > 0.0f ? v : 0.0f;
    }
    __syncthreads();

    const int lane = tid & 31;
    const int li = lane & 15;
    const int lh = lane >> 4;
    const int col = ((tid >> 5) << 4) + li;   // wave*16 + li, in [0,48)
    const bool ok = col < N_CLASS;

    v8f acc = {};
    for (int k = 0; k < N_HID; k += 4) {
        v2f a;
        a.x = ldsA[li * LDSPITCH + k + 2 * lh];
        a.y = ldsA[li * LDSPITCH + k + 2 * lh + 1];
        v2f b;
        b.x = ok ? W2[(long)(k + 2 * lh) * N_CLASS + col] : 0.0f;
        b.y = ok ? W2[(long)(k + 2 * lh + 1) * N_CLASS + col] : 0.0f;
        acc = __builtin_amdgcn_wmma_f32_16x16x4_f32(false, a, false, b, (short)0, acc, false, false);
    }
    if (ok) {
#pragma unroll
        for (int r = 0; r < 8; ++r) {
            int row = m0 + r + 8 * lh;
            L0[(long)row * N_CLASS + col] = acc[r];
        }
    }
}

// ---------------------------------------------------------------- SpMM (40 feats)
// one thread per (edge, float4 chunk): 10 chunks/edge; rows are 160B so chunks stay 16B-aligned
__global__ __launch_bounds__(256)
void SIGN_spmm40(const float* __restrict__ L0, const int* __restrict__ src,
                 const int* __restrict__ dst, const float* __restrict__ w,
                 float* __restrict__ OUT) {
    long t = (long)blockIdx.x * blockDim.x + threadIdx.x;
    if (t >= (long)N_EDGES * 10) return;
    int e = (int)(t / 10);
    int c = (int)(t - (long)e * 10) << 2;
    int s = src[e], d = dst[e];
    float we = w[e];
    const float4 v = *(const float4*)(L0 + (long)s * N_CLASS + c);
    float* o = OUT + (long)d * N_CLASS + c;
    atomicAdd(o + 0, we * v.x);
    atomicAdd(o + 1, we * v.y);
    atomicAdd(o + 2, we * v.z);
    atomicAdd(o + 3, we * v.w);
}

// ---------------------------------------------------------------- bias + log_softmax, in place
// one wave32 per row of 40: lane covers cols {lane, lane+32}
__global__ __launch_bounds__(256)
void SIGN_logsoftmax(float* __restrict__ OUT, const float* __restrict__ B2) {
    int row = blockIdx.x * 8 + (threadIdx.x >> 5);
    if (row >= N_NODES) return;
    int lane = threadIdx.x & 31;
    float* p = OUT + (long)row * N_CLASS;

    float v0 = p[lane] + B2[lane];                       // lane < 32 < 40 always valid
    bool has1 = (lane + 32) < N_CLASS;                   // lanes 0..7
    float v1 = has1 ? (p[lane + 32] + B2[lane + 32]) : -INFINITY;

    float m = fmaxf(v0, v1);
#pragma unroll
    for (int off = 16; off > 0; off >>= 1)
        m = fmaxf(m, __shfl_xor(m, off, 32));

    float s = expf(v0 - m) + (has1 ? expf(v1 - m) : 0.0f);
#pragma unroll
    for (int off = 16; off > 0; off >>= 1)
        s += __shfl_xor(s, off, 32);

    float lse = m + logf(s);
    p[lane] = v0 - lse;
    if (has1) p[lane + 32] = v1 - lse;
}

// ---------------------------------------------------------------- launch
extern "C" void kernel_launch(void* const* d_in, const int* in_sizes, int n_in,
                              void* d_out, int out_size, void* d_ws, size_t ws_size,
                              hipStream_t stream) {
    const float* x    = (const float*)d_in[0];
    const int*   esrc = (const int*)  d_in[1];
    const int*   edst = (const int*)  d_in[2];
    const float* ew   = (const float*)d_in[3];
    const float* W1   = (const float*)d_in[4];
    const float* b1   = (const float*)d_in[5];
    const float* W2   = (const float*)d_in[6];
    const float* b2   = (const float*)d_in[7];
    float* out = (float*)d_out;

    float* h0 = (float*)d_ws;                       // 100000*256 f32
    float* h1 = h0 + (size_t)N_NODES * N_HID;       // 100000*256 f32
    float* l0 = h1 + (size_t)N_NODES * N_HID;       // 100000*40  f32

    // zero the two atomic accumulators (h1, out)
    long nz = (long)N_NODES * N_HID;
    SIGN_zero_f32<<<(int)((nz + 255) / 256), 256, 0, stream>>>(h1, nz);
    long no = (long)N_NODES * N_CLASS;
    SIGN_zero_f32<<<(int)((no + 255) / 256), 256, 0, stream>>>(out, no);

    // layer 1
    SIGN_gemm1<<<N_NODES / 16, 256, 0, stream>>>(x, W1, h0);
    long t1 = (long)N_EDGES * 64;
    SIGN_spmm256<<<(int)((t1 + 255) / 256), 256, 0, stream>>>(h0, esrc, edst, ew, h1);

    // layer 2 (bias+relu fused into GEMM2's A staging)
    SIGN_gemm2<<<N_NODES / 16, 96, 0, stream>>>(h1, b1, W2, l0);
    long t2 = (long)N_EDGES * 10;
    SIGN_spmm40<<<(int)((t2 + 255) / 256), 256, 0, stream>>>(l0, esrc, edst, ew, out);

    // bias + log_softmax in place on d_out
    SIGN_logsoftmax<<<(N_NODES + 7) / 8, 256, 0, stream>>>(out, b2);
}